// WaveletPacketTransform_28956669509977
// MI455X (gfx1250) — compile-verified
//
#include <hip/hip_runtime.h>

typedef float v2f __attribute__((ext_vector_type(2)));
typedef float v8f __attribute__((ext_vector_type(8)));

typedef __attribute__((address_space(1))) int as1_int;   // global
typedef __attribute__((address_space(3))) int as3_int;   // LDS

#define XS_STRIDE 72      // 70-float rows padded to 72 dwords (keeps b64 loads 8B-aligned)
#define OS_STRIDE 260     // 256 floats per band + 4 pad (16B-aligned, de-conflicts banks)

#if defined(__has_builtin)
#  if __has_builtin(__builtin_amdgcn_global_load_async_to_lds_b32)
#    define USE_ASYNC_LDS 1
#  endif
#endif

__global__ __launch_bounds__(128)
void wpt2_db2_kernel(const float* __restrict__ x,
                     const float* __restrict__ dec_lo,
                     const float* __restrict__ dec_hi,
                     float* __restrict__ out) {
  __shared__ __align__(16) float Xs[70 * XS_STRIDE];   // 20160 B
  __shared__ __align__(16) float Os[16 * OS_STRIDE];   // 16640 B

  const int tl  = blockIdx.x;        // tile col (16 output cols)
  const int tk  = blockIdx.y;        // tile row (16 output rows)
  const int ch  = blockIdx.z;        // channel
  const int tid = threadIdx.x;
  const int lid = tid & 31;
  const int wv  = tid >> 5;          // wave 0..3
  const int half = (lid >> 4) & 1;   // K-half of the wave
  const int l15  = lid & 15;         // = A's M (output col l) = B's N (band)

  // ---------------- load 70x70 haloed input region (periodic wrap) ----------------
  const int r0 = (tk * 64 + 2048 - 6) & 2047;
  const int c0 = (tl * 64 + 2048 - 6) & 2047;
  const float* xc = x + (size_t)ch * (2048u * 2048u);
#ifdef USE_ASYNC_LDS
  for (int idx = tid; idx < 70 * 70; idx += 128) {
    int i  = idx / 70;
    int j  = idx - i * 70;
    int gr = (r0 + i) & 2047;
    int gc = (c0 + j) & 2047;
    __builtin_amdgcn_global_load_async_to_lds_b32(
        (as1_int*)(xc + (size_t)gr * 2048 + gc),
        (as3_int*)(Xs + i * XS_STRIDE + j),
        0, 0);
  }
#else
  for (int idx = tid; idx < 70 * 70; idx += 128) {
    int i  = idx / 70;
    int j  = idx - i * 70;
    int gr = (r0 + i) & 2047;
    int gc = (c0 + j) & 2047;
    Xs[i * XS_STRIDE + j] = xc[(size_t)gr * 2048 + gc];
  }
#endif

  // ---------------- per-lane composite 10-tap filters for band (lid&15) -----------
  // band b = 4*rf + cf; GRAY[f] = (level1 band a1, level2 band a2):
  //   f: 0->(0,0) 1->(0,1) 2->(1,1) 3->(1,0)  => a1=(f>>1)&1, a2=((f+1)>>1)&1
  // composite: g[m] = sum_{2*j2+j1==m} f2[j2]*f1[j1],  y[k]=sum_m g[m]*x[4k+9-m]
  float LO[4], HI[4];
#pragma unroll
  for (int j = 0; j < 4; ++j) { LO[j] = dec_lo[j]; HI[j] = dec_hi[j]; }

  const int rf = l15 >> 2;
  const int cf = l15 & 3;
  float gR[10], gC[10];
  {
    const int a1r = (rf >> 1) & 1, a2r = ((rf + 1) >> 1) & 1;
    const int a1c = (cf >> 1) & 1, a2c = ((cf + 1) >> 1) & 1;
#pragma unroll
    for (int m = 0; m < 10; ++m) { gR[m] = 0.f; gC[m] = 0.f; }
#pragma unroll
    for (int j2 = 0; j2 < 4; ++j2) {
      const float f2r = a2r ? HI[j2] : LO[j2];
      const float f2c = a2c ? HI[j2] : LO[j2];
#pragma unroll
      for (int j1 = 0; j1 < 4; ++j1) {
        const float f1r = a1r ? HI[j1] : LO[j1];
        const float f1c = a1c ? HI[j1] : LO[j1];
        gR[2 * j2 + j1] += f2r * f1r;
        gC[2 * j2 + j1] += f2c * f1c;
      }
    }
  }

  // ---------------- hoisted, chunk-invariant B fragments and A offsets -------------
  // lane's contraction pair: j0 = 4t + 2*half (even -> j0 and j0+1 share m)
  v2f Bfrag[25];
  int relOff[25];      // dword offset of A fragment relative to row 4q of Xs
#pragma unroll
  for (int t = 0; t < 25; ++t) {
    const int j0L = 4 * t,     mL = j0L / 10, uL = j0L % 10;
    const int j0H = 4 * t + 2, mH = j0H / 10, uH = j0H % 10;
    const float bxL = gR[mL] * gC[9 - uL], byL = gR[mL] * gC[8 - uL];
    const float bxH = gR[mH] * gC[9 - uH], byH = gR[mH] * gC[8 - uH];
    Bfrag[t].x = half ? bxH : bxL;
    Bfrag[t].y = half ? byH : byL;
    const int rL = (9 - mL) * XS_STRIDE + uL;
    const int rH = (9 - mH) * XS_STRIDE + uH;
    relOff[t] = (half ? rH : rL) + 4 * l15;     // even -> 8B-aligned b64 loads
  }

#ifdef USE_ASYNC_LDS
  asm volatile("s_wait_asynccnt 0x0" ::: "memory");
#endif
  __syncthreads();

  // ---------------- compute: Y[(k,l),band] = sum_j X[4k+9-m][4l+u]*gR[m]*gC[9-u] ---
  // GEMM per output row: M=16 (l), N=16 (band), K=100 -> 25 x V_WMMA_F32_16X16X4_F32.
  // Batch all 25 A loads into distinct regs first so ds latency overlaps WMMA issue.
#pragma unroll
  for (int qq = 0; qq < 4; ++qq) {
    const int q = wv * 4 + qq;     // output row k within the tile
    const float* base = Xs + 4 * q * XS_STRIDE;

    v2f A[25];
#pragma unroll
    for (int t = 0; t < 25; ++t)
      A[t] = *(const v2f*)(base + relOff[t]);

    v8f acc0 = {0.f, 0.f, 0.f, 0.f, 0.f, 0.f, 0.f, 0.f};
    v8f acc1 = {0.f, 0.f, 0.f, 0.f, 0.f, 0.f, 0.f, 0.f};
#pragma unroll
    for (int t = 0; t < 25; ++t) {
      if (t & 1)
        acc1 = __builtin_amdgcn_wmma_f32_16x16x4_f32(
            false, A[t], false, Bfrag[t], (short)0, acc1, false, false);
      else
        acc0 = __builtin_amdgcn_wmma_f32_16x16x4_f32(
            false, A[t], false, Bfrag[t], (short)0, acc0, false, false);
    }
    v8f acc = acc0 + acc1;

    // stage D: lane holds D[M = v+8*half][N = l15] = Y[row q][col M] of band l15
#pragma unroll
    for (int v = 0; v < 8; v += 2) {
      v2f p;
      p.x = acc[v];
      p.y = acc[v + 1];
      *(v2f*)(Os + l15 * OS_STRIDE + q * 16 + (v + 8 * half)) = p;
    }
  }

  __syncthreads();

  // ---------------- coalesced copy-out: 16 bands x 16x16 tile, float4 stores -------
  float* op = out + (size_t)ch * 16u * 512u * 512u;
#pragma unroll
  for (int it = 0; it < 8; ++it) {
    const int v4   = tid + it * 128;     // 0..1023 float4's
    const int e    = v4 * 4;
    const int band = e >> 8;
    const int k    = (e >> 4) & 15;
    const int l    = e & 15;
    const float4 val = *(const float4*)(Os + band * OS_STRIDE + k * 16 + l);
    float* dst = op + ((size_t)band * 512 + (tk * 16 + k)) * 512 + (tl * 16 + l);
    *(float4*)dst = val;
  }
}

extern "C" void kernel_launch(void* const* d_in, const int* in_sizes, int n_in,
                              void* d_out, int out_size, void* d_ws, size_t ws_size,
                              hipStream_t stream) {
  const float* x      = (const float*)d_in[0];
  const float* dec_lo = (const float*)d_in[1];
  const float* dec_hi = (const float*)d_in[2];
  float* out = (float*)d_out;

  const int C = in_sizes[0] / (2048 * 2048);   // 3
  dim3 grid(32, 32, C);
  dim3 block(128);
  hipLaunchKernelGGL(wpt2_db2_kernel, grid, block, 0, stream,
                     x, dec_lo, dec_hi, out);
}